// BSplineTransformation_5394478924596
// MI455X (gfx1250) — compile-verified
//
#include <hip/hip_runtime.h>
#include <cstdint>

namespace {

constexpr int IW  = 1024;
constexpr int IH  = 1024;
constexpr int HW  = IW * IH;        // 1,048,576
constexpr int NPLANES = 48;         // B*C = 16*3
constexpr int CP  = 35;             // control points per side
constexpr int CPN = CP * CP;        // 1225

// Faithful replica of _grid_sample_bilinear_border for one channel at one
// (gx, gy) "normalized" coordinate (the caller feeds whatever the reference
// feeds, including its un-normalized control-point coords — the clip makes
// that well-defined).
__device__ __forceinline__ float gs_bilinear_border(const float* __restrict__ img,
                                                    int ih, int iw,
                                                    float gx, float gy) {
  float ix = (gx + 1.0f) * 0.5f * (float)(iw - 1);
  float iy = (gy + 1.0f) * 0.5f * (float)(ih - 1);
  ix = fminf(fmaxf(ix, 0.0f), (float)(iw - 1));
  iy = fminf(fmaxf(iy, 0.0f), (float)(ih - 1));
  float ix0f = floorf(ix);
  float iy0f = floorf(iy);
  float wx = ix - ix0f;
  float wy = iy - iy0f;
  int ix0 = (int)ix0f;
  int iy0 = (int)iy0f;
  int ix1 = min(ix0 + 1, iw - 1);
  int iy1 = min(iy0 + 1, ih - 1);
  float v00 = img[iy0 * iw + ix0];
  float v01 = img[iy0 * iw + ix1];
  float v10 = img[iy1 * iw + ix0];
  float v11 = img[iy1 * iw + ix1];
  float top = v00 * (1.0f - wx) + v01 * wx;
  float bot = v10 * (1.0f - wx) + v11 * wx;
  return top * (1.0f - wy) + bot * wy;
}

// Kernel 1: deformation field (2, 1024, 1024).
// Stages the 2*35*35 f32 control-point table into LDS with CDNA5 async
// global->LDS copies (ASYNCcnt path), then each lane does 8 LDS bilinear taps.
__global__ __launch_bounds__(256) void bspline_def_field(
    const float* __restrict__ cp, float* __restrict__ dfield) {
  __shared__ float s_cp[2 * CPN];

  // CDNA5 async tensor-path staging: global -> LDS without touching VGPRs.
  // Generic pointers to LDS carry the wave-relative LDS byte offset in their
  // low 32 bits (aperture segmented addressing), so truncation yields VDST.
  for (int i = (int)threadIdx.x; i < 2 * CPN; i += 256) {
    const float* src = cp + i;
    uint32_t lds_off = (uint32_t)(uintptr_t)(&s_cp[i]);
    asm volatile("global_load_async_to_lds_b32 %0, %1, off"
                 :: "v"(lds_off), "v"(src)
                 : "memory");
  }
  asm volatile("s_wait_asynccnt 0" ::: "memory");
  __syncthreads();

  int p = (int)blockIdx.x * 256 + (int)threadIdx.x;   // grid exactly covers HW
  int ypix = p >> 10;
  int xpix = p & (IW - 1);

  // Reference: cp_x = gx*(cp-1)/2 + (cp-1)/2  ==  x * (cp-1)/(W-1)
  float cpxc = (float)xpix * ((float)(CP - 1) / (float)(IW - 1));
  float cpyc = (float)ypix * ((float)(CP - 1) / (float)(IH - 1));

  float d0 = gs_bilinear_border(s_cp,       CP, CP, cpxc, cpyc); // channel 0 (y-disp)
  float d1 = gs_bilinear_border(s_cp + CPN, CP, CP, cpxc, cpyc); // channel 1 (x-disp)

  // Regular (temporal) stores: kernel 2 re-reads this 8 MB field from L2.
  dfield[p]      = d0;
  dfield[HW + p] = d1;
}

// Kernel 2: warp all 48 planes. One lane per output pixel per plane.
// blockIdx.y = plane (b*3+c). Gather 4 taps, blend, NT-store the result so
// the streaming 201 MB output does not evict the reusable input from L2.
__global__ __launch_bounds__(256) void bspline_warp(
    const float* __restrict__ img, const float* __restrict__ dfield,
    float* __restrict__ out) {
  int p = (int)blockIdx.x * 256 + (int)threadIdx.x;
  int plane = (int)blockIdx.y;
  int ypix = p >> 10;
  int xpix = p & (IW - 1);

  float gx = fmaf((float)xpix, 2.0f / (float)(IW - 1), -1.0f);
  float gy = fmaf((float)ypix, 2.0f / (float)(IH - 1), -1.0f);

  // samp = (gx + def[1], gy + def[0])  — matches reference channel order.
  float sx = gx + dfield[HW + p];
  float sy = gy + dfield[p];

  const float* base = img + (size_t)plane * (size_t)HW;
  float r = gs_bilinear_border(base, IH, IW, sx, sy);

  __builtin_nontemporal_store(r, &out[(size_t)plane * (size_t)HW + p]);
}

} // namespace

extern "C" void kernel_launch(void* const* d_in, const int* in_sizes, int n_in,
                              void* d_out, int out_size, void* d_ws, size_t ws_size,
                              hipStream_t stream) {
  (void)in_sizes; (void)n_in; (void)out_size; (void)d_ws; (void)ws_size;

  const float* x  = (const float*)d_in[0];   // (16,3,1024,1024) f32
  const float* cp = (const float*)d_in[1];   // (1,2,35,35)      f32

  float* out    = (float*)d_out;                       // transformed, 48*HW floats
  float* dfield = out + (size_t)NPLANES * (size_t)HW;  // deformation field, 2*HW floats

  // Kernel 1: 1M pixels -> 4096 blocks of 256 (8 wave32 each).
  bspline_def_field<<<HW / 256, 256, 0, stream>>>(cp, dfield);

  // Kernel 2: (pixels/256, planes).
  dim3 grid(HW / 256, NPLANES);
  bspline_warp<<<grid, 256, 0, stream>>>(x, dfield, out);
}